// GraphAutoencoder_66623532695725
// MI455X (gfx1250) — compile-verified
//
#include <hip/hip_runtime.h>
#include <hip/hip_bf16.h>
#include <math.h>

typedef float v2f __attribute__((ext_vector_type(2)));
typedef float v8f __attribute__((ext_vector_type(8)));

#define TPB 256

// ---------------- utility kernels ----------------

__global__ __launch_bounds__(TPB) void k_fill(float* __restrict__ p, float val, int n) {
    int i = blockIdx.x * TPB + threadIdx.x;
    if (i < n) p[i] = val;
}

__global__ __launch_bounds__(TPB) void k_deg_accum(const int* __restrict__ dst,
                                                   float* __restrict__ deg, int E) {
    int e = blockIdx.x * TPB + threadIdx.x;
    if (e < E) atomicAdd(&deg[dst[e]], 1.0f);
}

__global__ __launch_bounds__(TPB) void k_rsqrt_inplace(float* __restrict__ p, int n) {
    int i = blockIdx.x * TPB + threadIdx.x;
    if (i < n) p[i] = rsqrtf(p[i]);
}

// ---------------- WMMA GEMM: C[M,NC] = A[M,K] @ B[K,NC] ----------------
// f32 V_WMMA 16x16x4. One wave computes one 16x16 output tile.
// A-frag (16x4):  lane%16 = M row, K = k0 + 2*(lane/16) + vgpr   -> v2f
// B-frag (4x16):  lane%16 = N col, K = k0 + 2*(lane/16) + vgpr   -> v2f
// C/D  (16x16):   lane%16 = N col, M row = vgpr + 8*(lane/16)    -> v8f
// Requires M % 16 == 0, NC % 16 == 0 (N=100000 nodes -> 6250 M-tiles exactly).
template <int K, int NC>
__global__ __launch_bounds__(TPB) void k_gemm_wmma(const float* __restrict__ A,
                                                   const float* __restrict__ B,
                                                   float* __restrict__ C,
                                                   int totalTiles) {
    int wid  = (blockIdx.x * TPB + threadIdx.x) >> 5;   // uniform within a wave
    int lane = threadIdx.x & 31;
    if (wid >= totalTiles) return;                      // whole wave exits together

    constexpr int NT = NC / 16;
    int mt   = wid / NT;
    int nt   = wid - mt * NT;
    int half = lane >> 4;     // 0: K pair {0,1}, 1: K pair {2,3}
    int l16  = lane & 15;

    const float* arow = A + (size_t)(mt * 16 + l16) * K;
    int bcol = nt * 16 + l16;

    v8f acc = {0.f, 0.f, 0.f, 0.f, 0.f, 0.f, 0.f, 0.f};
#pragma unroll
    for (int k0 = 0; k0 < K; k0 += 4) {
        int ka = k0 + 2 * half;
        v2f a = *(const v2f*)(arow + ka);      // K=ka, ka+1 (8B aligned: ka even)
        v2f b;
        b.x = B[(size_t)ka * NC + bcol];
        b.y = B[(size_t)(ka + 1) * NC + bcol];
        acc = __builtin_amdgcn_wmma_f32_16x16x4_f32(
            /*neg_a=*/false, a, /*neg_b=*/false, b,
            /*c_mod=*/(short)0, acc, /*reuse_a=*/false, /*reuse_b=*/false);
    }
#pragma unroll
    for (int v = 0; v < 8; v++) {
        C[(size_t)(mt * 16 + v + 8 * half) * NC + bcol] = acc[v];
    }
}

// ---------------- edge scatter: agg[dst] += feat[src] * dinv[src]*dinv[dst] ----
// One wave per edge: gather is a contiguous C*4-byte wave-coalesced read (L2-resident),
// scatter is C non-returning global_atomic_add_f32 (contiguous per wave).
template <int C>
__global__ __launch_bounds__(TPB) void k_scatter(const float* __restrict__ feat,
                                                 const int* __restrict__ src,
                                                 const int* __restrict__ dst,
                                                 const float* __restrict__ dinv,
                                                 float* __restrict__ agg, int E) {
    long long gt = (long long)blockIdx.x * TPB + threadIdx.x;
    int e    = (int)(gt >> 5);
    int lane = (int)(gt & 31);
    if (e >= E) return;
    int s = src[e], d = dst[e];
    float norm = dinv[s] * dinv[d];
    const float* fs = feat + (size_t)s * C;
    float*       ad = agg  + (size_t)d * C;
#pragma unroll
    for (int c = 0; c < C; c += 32) {
        atomicAdd(&ad[c + lane], fs[c + lane] * norm);
    }
}

// ---------------- fused epilogues ----------------
// h = relu( BN( agg1 + xw*dinv^2 + b1 ) ), written in place over agg1.
__global__ __launch_bounds__(TPB) void k_post1(float* __restrict__ h,
                                               const float* __restrict__ xw,
                                               const float* __restrict__ dinv,
                                               const float* __restrict__ b1,
                                               const float* __restrict__ gamma,
                                               const float* __restrict__ beta,
                                               const float* __restrict__ mean,
                                               const float* __restrict__ var,
                                               int total) {
    int i = blockIdx.x * TPB + threadIdx.x;
    if (i >= total) return;
    int node = i >> 6;          // 64 channels
    int c    = i & 63;
    float di = dinv[node];
    float v  = h[i] + xw[i] * di * di + b1[c];
    v = (v - mean[c]) * rsqrtf(var[c] + 1e-5f) * gamma[c] + beta[c];
    h[i] = v > 0.f ? v : 0.f;
}

// out = out + hw*dinv^2 + b2 (in place over atomically-accumulated d_out)
__global__ __launch_bounds__(TPB) void k_post2(float* __restrict__ out,
                                               const float* __restrict__ hw,
                                               const float* __restrict__ dinv,
                                               const float* __restrict__ b2,
                                               int total) {
    int i = blockIdx.x * TPB + threadIdx.x;
    if (i >= total) return;
    int node = i >> 5;          // 32 channels
    int c    = i & 31;
    float di = dinv[node];
    out[i] = out[i] + hw[i] * di * di + b2[c];
}

// ---------------- launcher ----------------

static inline int cdiv_ll(long long a, long long b) { return (int)((a + b - 1) / b); }

extern "C" void kernel_launch(void* const* d_in, const int* in_sizes, int n_in,
                              void* d_out, int out_size, void* d_ws, size_t ws_size,
                              hipStream_t stream) {
    const float* x     = (const float*)d_in[0];
    const int*   ei    = (const int*)d_in[1];   // integer input -> int32 per harness
    const float* W1    = (const float*)d_in[2];
    const float* b1    = (const float*)d_in[3];
    const float* gamma = (const float*)d_in[4];
    const float* beta  = (const float*)d_in[5];
    const float* rmean = (const float*)d_in[6];
    const float* rvar  = (const float*)d_in[7];
    const float* W2    = (const float*)d_in[8];
    const float* b2    = (const float*)d_in[9];

    const int N = in_sizes[0] / 128;    // IN_CH = 128
    const int E = in_sizes[1] / 2;      // edge_index is [2, E]
    const int* src = ei;
    const int* dst = ei + E;
    float* out = (float*)d_out;

    // workspace layout (floats): dinv[N] | xw[N*64] | agg1/h[N*64] | hw[N*32]
    float* ws   = (float*)d_ws;
    float* dinv = ws;
    float* xw   = dinv + N;
    float* agg1 = xw + (size_t)N * 64;
    float* hw   = agg1 + (size_t)N * 64;

    // --- degree (in-deg + self loop) -> dinv = rsqrt(deg) ---
    k_fill<<<cdiv_ll(N, TPB), TPB, 0, stream>>>(dinv, 1.0f, N);
    k_deg_accum<<<cdiv_ll(E, TPB), TPB, 0, stream>>>(dst, dinv, E);
    k_rsqrt_inplace<<<cdiv_ll(N, TPB), TPB, 0, stream>>>(dinv, N);

    // --- layer 1 ---
    k_fill<<<cdiv_ll((long long)N * 64, TPB), TPB, 0, stream>>>(agg1, 0.0f, N * 64);
    int tiles1 = (N / 16) * (64 / 16);
    k_gemm_wmma<128, 64><<<cdiv_ll((long long)tiles1 * 32, TPB), TPB, 0, stream>>>(
        x, W1, xw, tiles1);
    k_scatter<64><<<cdiv_ll((long long)E * 32, TPB), TPB, 0, stream>>>(
        xw, src, dst, dinv, agg1, E);
    k_post1<<<cdiv_ll((long long)N * 64, TPB), TPB, 0, stream>>>(
        agg1, xw, dinv, b1, gamma, beta, rmean, rvar, N * 64);

    // --- layer 2 ---
    int tiles2 = (N / 16) * (32 / 16);
    k_gemm_wmma<64, 32><<<cdiv_ll((long long)tiles2 * 32, TPB), TPB, 0, stream>>>(
        agg1, W2, hw, tiles2);
    k_fill<<<cdiv_ll((long long)N * 32, TPB), TPB, 0, stream>>>(out, 0.0f, N * 32);
    k_scatter<32><<<cdiv_ll((long long)E * 32, TPB), TPB, 0, stream>>>(
        hw, src, dst, dinv, out, E);
    k_post2<<<cdiv_ll((long long)N * 32, TPB), TPB, 0, stream>>>(
        out, hw, dinv, b2, N * 32);
}